// AttentionSE3_74406013435998
// MI455X (gfx1250) — compile-verified
//
#include <hip/hip_runtime.h>
#include <hip/hip_bf16.h>
#include <math.h>

typedef __attribute__((ext_vector_type(8))) float v8f;
typedef __attribute__((ext_vector_type(4))) float v4f;
typedef __attribute__((ext_vector_type(2))) float v2f;

#define HNUM 8
#define DK   16
#define SCALE 0.08838834764831845f  // 1/sqrt(128)

// ---- order-preserving float<->uint encoding for atomic max ----
__device__ __forceinline__ unsigned enc_f32(float f) {
  unsigned u = __float_as_uint(f);
  return (u & 0x80000000u) ? ~u : (u | 0x80000000u);
}
__device__ __forceinline__ float dec_f32(unsigned u) {
  return (u & 0x80000000u) ? __uint_as_float(u ^ 0x80000000u)
                           : __uint_as_float(~u);
}

// ---- Stage 1: fuse query0 [N,32,1] + query1 [N,32,3] -> qbuf [N,128] ----
__global__ void build_q_kernel(const float* __restrict__ q0,
                               const float* __restrict__ q1,
                               float* __restrict__ qbuf, int N) {
  int t = blockIdx.x * blockDim.x + threadIdx.x;      // one thread per (n, c)
  if (t >= N * 32) return;
  int n = t >> 5, c = t & 31;
  v4f v;
  v[0] = q0[(size_t)n * 32 + c];
  const float* p1 = q1 + ((size_t)n * 32 + c) * 3;
  v[1] = p1[0]; v[2] = p1[1]; v[3] = p1[2];
  *(v4f*)(qbuf + (size_t)n * 128 + (size_t)c * 4) = v;
}

// ---- Stage 2: logits via chained V_WMMA_F32_16X16X4_F32 (exact f32 math) ----
// Tile = 16 edges x 1 head; 4 WMMAs chain K=16. A 16x4 f32 layout (ISA 7.12.2):
//   lane<16: M=lane,   VGPR0=K(4i), VGPR1=K(4i+1)
//   lane>=16: M=lane-16, VGPR0=K(4i+2), VGPR1=K(4i+3)
// B 4x16 mirrors it (K rows striped across VGPRs, N=lane columns).
// D's diagonal = per-edge logit: D[m,m] at (VGPR m, lane m) for m<8,
// (VGPR m-8, lane m+16) for m>=8. Extracted via a per-lane LDS bounce.
__global__ void logits_wmma_kernel(const float* __restrict__ key,
                                   const float* __restrict__ qbuf,
                                   const long long* __restrict__ dst,
                                   float* __restrict__ wbuf,
                                   unsigned* __restrict__ segmax,
                                   int E, int ntiles) {
  __shared__ float lds_c[256 * 8];                    // 8 floats per thread
  int wave = blockIdx.x * (blockDim.x >> 5) + (threadIdx.x >> 5);
  int lane = threadIdx.x & 31;
  if (wave >= ntiles) return;                         // wave-uniform exit
  int base = wave * 16;

  if (base + 16 <= E) {                               // full tile (EXEC all-1s at WMMA)
    bool hi = lane >= 16;
    int  l  = lane & 15;
    int  erow = base + l;
    const float* kp = key + (size_t)erow * 128 + (hi ? 2 : 0);
    long long dn = dst[erow];
    const float* qp = qbuf + (size_t)dn * 128 + (hi ? 2 : 0);
    if (erow + 16 < E)                                // global_prefetch_b8: next key tile
      __builtin_prefetch(key + (size_t)(erow + 16) * 128, 0, 0);
    float* myc = lds_c + (size_t)threadIdx.x * 8;

    int j = hi ? (lane - 24) : lane;                  // diag accumulator index (writers)
    bool writer = (lane < 8) || (lane >= 24);

#pragma unroll
    for (int h = 0; h < HNUM; ++h) {
      v8f c = {};
#pragma unroll
      for (int i = 0; i < 4; ++i) {                   // chain K = 16 in steps of 4
        v2f a = *(const v2f*)(kp + h * DK + i * 4);
        v2f b = *(const v2f*)(qp + h * DK + i * 4);
        c = __builtin_amdgcn_wmma_f32_16x16x4_f32(
            false, a, false, b, (short)0, c, false, false);
      }
      *(v8f*)myc = c;                                 // ds_store: own 32B slot
      if (writer) {
        float lg = myc[j] * SCALE;                    // ds_load_b32, in-order w/ store
        int e = base + l;                             // l == diagonal row m
        wbuf[(size_t)e * HNUM + h] = lg;
        atomicMax(&segmax[(size_t)dn * HNUM + h], enc_f32(lg));
      }
    }
  } else {                                            // tail tile: scalar path
    int rem = E - base;
    for (int p = lane; p < rem * HNUM; p += 32) {
      int e = base + (p >> 3), h = p & 7;
      const float* kp = key + (size_t)e * 128 + h * DK;
      long long dn = dst[e];
      const float* qp = qbuf + (size_t)dn * 128 + h * DK;
      float s = 0.f;
#pragma unroll
      for (int t = 0; t < DK; ++t) s += kp[t] * qp[t];
      s *= SCALE;
      wbuf[(size_t)e * HNUM + h] = s;
      atomicMax(&segmax[(size_t)dn * HNUM + h], enc_f32(s));
    }
  }
}

// ---- Stage 3: expw = exp(logit - segmax[dst]); segsum += expw ----
__global__ void exp_sum_kernel(const long long* __restrict__ dst,
                               float* __restrict__ wbuf,
                               const unsigned* __restrict__ segmax,
                               float* __restrict__ segsum, int E) {
  int t = blockIdx.x * blockDim.x + threadIdx.x;      // one thread per (e, h)
  if (t >= E * HNUM) return;
  int e = t >> 3, h = t & 7;
  long long n = dst[e];
  float m = dec_f32(segmax[(size_t)n * HNUM + h]);
  float w = expf(wbuf[t] - m);
  wbuf[t] = w;
  atomicAdd(&segsum[(size_t)n * HNUM + h], w);
}

// ---- Stage 4: out[dst] += (expw/segsum[dst]) * value ----
__global__ void scatter_kernel(const float* __restrict__ value,
                               const long long* __restrict__ dst,
                               const float* __restrict__ wbuf,
                               const float* __restrict__ segsum,
                               float* __restrict__ out, int E) {
  int t = blockIdx.x * blockDim.x + threadIdx.x;      // 32 lanes per edge
  if (t >= E * 32) return;
  int e = t >> 5, c = t & 31, h = c >> 2;
  long long n = dst[e];
  float a = wbuf[(size_t)e * HNUM + h] / segsum[(size_t)n * HNUM + h];
  const float* vp = value + (size_t)e * 96 + (size_t)c * 3;
  float*       op = out   + (size_t)n * 96 + (size_t)c * 3;
  atomicAdd(op + 0, a * vp[0]);
  atomicAdd(op + 1, a * vp[1]);
  atomicAdd(op + 2, a * vp[2]);
}

extern "C" void kernel_launch(void* const* d_in, const int* in_sizes, int n_in,
                              void* d_out, int out_size, void* d_ws, size_t ws_size,
                              hipStream_t stream) {
  const float*     value = (const float*)d_in[0];      // [E,32,3]
  const float*     key   = (const float*)d_in[1];      // [E,128]
  const float*     q0    = (const float*)d_in[2];      // [N,32,1]
  const float*     q1    = (const float*)d_in[3];      // [N,32,3]
  const long long* eidx  = (const long long*)d_in[4];  // [2,E] int64
  float* out = (float*)d_out;                          // [N,32,3]

  const int E = in_sizes[1] / 128;
  const int N = in_sizes[2] / 32;
  const long long* dst = eidx + E;                     // edge_index[1]

  // workspace layout
  char* ws = (char*)d_ws;
  float*    qbuf   = (float*)ws;                               // N*128
  float*    wbuf   = (float*)(ws + (size_t)N * 128 * 4);       // E*8
  unsigned* segmax = (unsigned*)(ws + ((size_t)N * 128 + (size_t)E * 8) * 4);  // N*8
  float*    segsum = (float*)(ws + ((size_t)N * 128 + (size_t)E * 8 + (size_t)N * 8) * 4);

  hipMemsetAsync(out,    0, (size_t)out_size * sizeof(float), stream);
  hipMemsetAsync(segmax, 0, (size_t)N * HNUM * 4, stream);     // 0 <= enc(any float)
  hipMemsetAsync(segsum, 0, (size_t)N * HNUM * 4, stream);

  { // stage 1: fused query
    int total = N * 32, blk = 256;
    build_q_kernel<<<(total + blk - 1) / blk, blk, 0, stream>>>(q0, q1, qbuf, N);
  }
  { // stage 2: WMMA logits + segment max
    int ntiles = (E + 15) / 16;
    int blk = 256;                                     // 8 waves -> 8 tiles per block
    int grid = (ntiles + 7) / 8;
    logits_wmma_kernel<<<grid, blk, 0, stream>>>(key, qbuf, dst, wbuf, segmax, E, ntiles);
  }
  { // stage 3: exp + segment sum
    int total = E * HNUM, blk = 256;
    exp_sum_kernel<<<(total + blk - 1) / blk, blk, 0, stream>>>(dst, wbuf, segmax, segsum, E);
  }
  { // stage 4: weighted scatter
    long long total = (long long)E * 32; int blk = 256;
    scatter_kernel<<<(int)((total + blk - 1) / blk), blk, 0, stream>>>(
        value, dst, wbuf, segsum, out, E);
  }
}